// GATRouting2Hop_32375463477763
// MI455X (gfx1250) — compile-verified
//
#include <hip/hip_runtime.h>

typedef float v2f __attribute__((ext_vector_type(2)));
typedef float v8f __attribute__((ext_vector_type(8)));

#define HID 256
#define NHEAD 8
#define CH 32
#define NEG_SLOPE 0.2f
#define LN_EPS 1e-5f
#define EPB 4  // edges per block in aggregation

// Async global->LDS 128-bit copy (gfx1250). Tracked by ASYNCcnt.
// LDS operand = 32-bit LDS offset (low 32 bits of generic shared pointer;
// aperture bits are [63:32] per CDNA5 flat addressing).
__device__ __forceinline__ void async_copy_b128(const float* gsrc, float* lds_dst)
{
  unsigned loff = (unsigned)(unsigned long long)lds_dst;
  asm volatile("global_load_async_to_lds_b128 %0, %1, off"
               :: "v"(loff), "v"(gsrc) : "memory");
}
__device__ __forceinline__ void wait_asynccnt0()
{
  asm volatile("s_wait_asynccnt 0x0" ::: "memory");
}

// ---------------------------------------------------------------------------
// C[M,256] = A[M,K] @ B[K,256] (+ optional bias), full fp32 via WMMA 16x16x4.
// Block: 256 threads (8 waves). grid = (M/16, 2); wave w owns ntile = gy*8+w.
// A tile (16 x K) staged in LDS by the async tensor path; B read through L2.
// ---------------------------------------------------------------------------
__global__ __launch_bounds__(256) void k_gemm_f32wmma(
    const float* __restrict__ A, const float* __restrict__ B,
    const float* __restrict__ bias, float* __restrict__ Cmat,
    int M, int K)
{
  __shared__ float As[16 * 256];                 // supports K up to 256 (16 KB)
  const int tid   = threadIdx.x;
  const int mbase = blockIdx.x << 4;
  const int k4    = K >> 2;

  for (int i = tid; i < 16 * k4; i += 256) {     // cooperative 16xK tile load
    int row = i / k4, col4 = i - row * k4;
    int gr  = mbase + row;
    float* dst = &As[(row * k4 + col4) * 4];
    if (gr < M) {
      async_copy_b128(A + (size_t)gr * K + 4 * col4, dst);
    } else {
      float4 z; z.x = z.y = z.z = z.w = 0.f;
      *(float4*)dst = z;
    }
  }
  wait_asynccnt0();
  __syncthreads();

  const int wave  = tid >> 5;
  const int lane  = tid & 31;
  const int khalf = lane >> 4;                   // lane half selects K pair
  const int l16   = lane & 15;
  const int ntile = (blockIdx.y << 3) + wave;
  const int ncol  = (ntile << 4) + l16;

  v8f acc = {};
  for (int k = 0; k < K; k += 4) {
    v2f a, b;
    // A 16x4 f32 layout: lane holds row M=lane%16, K = k + 2*(lane/16) + {0,1}
    a.x = As[l16 * K + k + 2 * khalf + 0];
    a.y = As[l16 * K + k + 2 * khalf + 1];
    // B 4x16 f32 layout (mirror): lane holds col N=lane%16, same K split
    b.x = B[(size_t)(k + 2 * khalf + 0) * HID + ncol];
    b.y = B[(size_t)(k + 2 * khalf + 1) * HID + ncol];
    acc = __builtin_amdgcn_wmma_f32_16x16x4_f32(false, a, false, b,
                                                (short)0, acc, false, false);
  }

  const float bv = bias ? bias[ncol] : 0.f;
  const int rbase = mbase + (khalf << 3);        // C: VGPR g -> row g + 8*khalf
  #pragma unroll
  for (int g = 0; g < 8; ++g) {
    int r = rbase + g;
    if (r < M) Cmat[(size_t)r * HID + ncol] = acc[g] + bv;
  }
}

// ---------------------------------------------------------------------------
// Per-node attention logits: al_src/al_dst[n,h] = sum_c h[n,h,c] * att[h,c]
// ---------------------------------------------------------------------------
__global__ __launch_bounds__(256) void k_node_alpha(
    const float* __restrict__ hbuf,
    const float* __restrict__ asrc, const float* __restrict__ adst,
    float* __restrict__ alsrc, float* __restrict__ aldst, int Nn)
{
  int t = blockIdx.x * 256 + threadIdx.x;
  if (t >= Nn * NHEAD) return;
  int n = t >> 3, h = t & 7;
  const float* hp = hbuf + (size_t)n * HID + h * CH;
  const float* as = asrc + h * CH;
  const float* ad = adst + h * CH;
  float s1 = 0.f, s2 = 0.f;
  #pragma unroll
  for (int c = 0; c < CH; ++c) { float v = hp[c]; s1 += v * as[c]; s2 += v * ad[c]; }
  alsrc[t] = s1; aldst[t] = s2;
}

// ae[h] = sum_c We[0, h*32+c] * a_edge[h,c]  (edge term collapses to ew*ae[h])
__global__ void k_ae(const float* __restrict__ We, const float* __restrict__ aedge,
                     float* __restrict__ ae)
{
  int h = threadIdx.x;
  if (h >= NHEAD) return;
  float s = 0.f;
  #pragma unroll
  for (int c = 0; c < CH; ++c) s += We[h * CH + c] * aedge[h * CH + c];
  ae[h] = s;
}

__global__ void k_fill(float* __restrict__ p, float v, int n)
{
  int i = blockIdx.x * 256 + threadIdx.x;
  if (i < n) p[i] = v;
}

__device__ inline void atomicMaxF(float* addr, float val)
{
  unsigned int* ua = (unsigned int*)addr;
  unsigned int old = *ua;
  while (__uint_as_float(old) < val) {
    unsigned int assumed = old;
    old = atomicCAS(ua, assumed, __float_as_uint(val));
    if (old == assumed) break;
  }
}

// ---------------------------------------------------------------------------
// Edge pass A: alpha[e,h] = leaky_relu(al_src[src]+al_dst[dst]+ew*ae[h]);
//              m[dst,h] = max(...)
// ---------------------------------------------------------------------------
__global__ __launch_bounds__(256) void k_edge_max(
    const long long* __restrict__ ei,
    const float* __restrict__ alsrc, const float* __restrict__ aldst,
    const float* __restrict__ ew, const float* __restrict__ ae,
    float* __restrict__ alpha, float* __restrict__ m, int E)
{
  int t = blockIdx.x * 256 + threadIdx.x;
  if (t >= E * NHEAD) return;
  int e = t >> 3, h = t & 7;
  int s = (int)ei[e];
  int d = (int)ei[(size_t)E + e];
  float a = alsrc[s * NHEAD + h] + aldst[d * NHEAD + h] + ew[e] * ae[h];
  a = (a > 0.f) ? a : NEG_SLOPE * a;
  alpha[t] = a;
  atomicMaxF(&m[d * NHEAD + h], a);
}

// ---------------------------------------------------------------------------
// Edge pass B: p = exp(alpha - m[dst]); denom[dst,h] += p;
//              agg[dst,:] += p * h[src,:]   (unnormalized; divide per node later)
// One wave = one head = 32 contiguous channels -> coalesced 128B gather/scatter.
// Next edge's source row is prefetched (global_prefetch_b8) to hide L2 latency.
// ---------------------------------------------------------------------------
__global__ __launch_bounds__(256) void k_edge_agg(
    const long long* __restrict__ ei,
    const float* __restrict__ alpha, const float* __restrict__ m,
    const float* __restrict__ hsrc,
    float* __restrict__ denom, float* __restrict__ agg, int E)
{
  const int h = threadIdx.x >> 5;        // wave index == head
  const int lane = threadIdx.x & 31;
  const int e0 = blockIdx.x * EPB;
  for (int i = 0; i < EPB; ++i) {
    int e = e0 + i;
    if (e >= E) return;
    if (i + 1 < EPB && e + 1 < E) {
      int sn = (int)ei[e + 1];
      __builtin_prefetch(&hsrc[(size_t)sn * HID + h * CH + lane], 0, 0);
    }
    int s = (int)ei[e];
    int d = (int)ei[(size_t)E + e];
    float p = 0.f;
    if (lane == 0) {
      p = __expf(alpha[e * NHEAD + h] - m[d * NHEAD + h]);
      atomicAdd(&denom[d * NHEAD + h], p);
    }
    p = __shfl(p, 0, 32);
    float v = hsrc[(size_t)s * HID + h * CH + lane];
    atomicAdd(&agg[(size_t)d * HID + h * CH + lane], p * v);
  }
}

// ---------------------------------------------------------------------------
// Per-node epilogue: normalize, head-mean (broadcast across heads) + bias,
// optional ReLU, add residual, LayerNorm. One block (256 thr) per node.
// ---------------------------------------------------------------------------
__global__ __launch_bounds__(256) void k_epilogue(
    const float* __restrict__ agg, const float* __restrict__ denom,
    const float* __restrict__ bias, const float* __restrict__ resid,
    const float* __restrict__ ln_g, const float* __restrict__ ln_b,
    float* __restrict__ out, int do_relu)
{
  __shared__ float sv[HID];
  __shared__ float sm[CH];
  __shared__ float red[16];
  __shared__ float mu_s, rstd_s;
  const int n = blockIdx.x, t = threadIdx.x;
  const int h = t >> 5, c = t & 31;

  float v = agg[(size_t)n * HID + t] / (denom[n * NHEAD + h] + 1e-16f);
  sv[t] = v;
  __syncthreads();
  if (t < CH) {                                  // mean over 8 heads per channel
    float s = 0.f;
    #pragma unroll
    for (int k = 0; k < NHEAD; ++k) s += sv[k * CH + t];
    sm[t] = s * (1.f / NHEAD);
  }
  __syncthreads();

  float d = sm[c] + bias[t];
  if (do_relu) d = fmaxf(d, 0.f);
  float pre = resid[(size_t)n * HID + t] + d;

  float s = pre, ss = pre * pre;                 // wave32 tree reduction
  #pragma unroll
  for (int off = 16; off > 0; off >>= 1) {
    s  += __shfl_down(s,  off, 32);
    ss += __shfl_down(ss, off, 32);
  }
  if ((t & 31) == 0) { red[h] = s; red[8 + h] = ss; }
  __syncthreads();
  if (t == 0) {
    float S = 0.f, SS = 0.f;
    #pragma unroll
    for (int k = 0; k < NHEAD; ++k) { S += red[k]; SS += red[8 + k]; }
    float mu = S * (1.f / HID);
    mu_s = mu;
    rstd_s = rsqrtf(SS * (1.f / HID) - mu * mu + LN_EPS);
  }
  __syncthreads();
  out[(size_t)n * HID + t] = (pre - mu_s) * rstd_s * ln_g[t] + ln_b[t];
}

// ---------------------------------------------------------------------------
extern "C" void kernel_launch(void* const* d_in, const int* in_sizes, int n_in,
                              void* d_out, int out_size, void* d_ws, size_t ws_size,
                              hipStream_t stream)
{
  const float*     x     = (const float*)d_in[0];
  const long long* ei    = (const long long*)d_in[1];
  const float*     ew    = (const float*)d_in[2];
  const float*     W1    = (const float*)d_in[3];
  const float*     asrc1 = (const float*)d_in[4];
  const float*     adst1 = (const float*)d_in[5];
  const float*     We1   = (const float*)d_in[6];
  const float*     aed1  = (const float*)d_in[7];
  const float*     b1    = (const float*)d_in[8];
  const float*     g1    = (const float*)d_in[9];
  const float*     be1   = (const float*)d_in[10];
  const float*     W2    = (const float*)d_in[11];
  const float*     asrc2 = (const float*)d_in[12];
  const float*     adst2 = (const float*)d_in[13];
  const float*     We2   = (const float*)d_in[14];
  const float*     aed2  = (const float*)d_in[15];
  const float*     b2    = (const float*)d_in[16];
  const float*     g2    = (const float*)d_in[17];
  const float*     be2   = (const float*)d_in[18];
  const float*     resW  = (const float*)d_in[19];
  const float*     resb  = (const float*)d_in[20];

  const int IN_DIM = 128;
  const int N = in_sizes[0] / IN_DIM;
  const int E = in_sizes[2];

  float* ws = (float*)d_ws;
  size_t off = 0;
  float* h_buf = ws + off; off += (size_t)N * HID;    // h1, then h2
  float* r_buf = ws + off; off += (size_t)N * HID;    // res, then h_mid (in place)
  float* agg   = ws + off; off += (size_t)N * HID;
  float* alpha = ws + off; off += (size_t)E * NHEAD;
  float* alsrc = ws + off; off += (size_t)N * NHEAD;
  float* aldst = ws + off; off += (size_t)N * NHEAD;
  float* mbuf  = ws + off; off += (size_t)N * NHEAD;
  float* dbuf  = ws + off; off += (size_t)N * NHEAD;
  float* aeb   = ws + off; off += 8;

  dim3 gemmGrid((N + 15) / 16, 2);
  const int nh  = N * NHEAD;
  const int eh  = E * NHEAD;
  const int nch = N * HID;

  // ---------------- Layer 1 ----------------
  k_gemm_f32wmma<<<gemmGrid, 256, 0, stream>>>(x, W1,  nullptr, h_buf, N, IN_DIM);
  k_gemm_f32wmma<<<gemmGrid, 256, 0, stream>>>(x, resW, resb,   r_buf, N, IN_DIM);
  k_node_alpha<<<(nh + 255) / 256, 256, 0, stream>>>(h_buf, asrc1, adst1, alsrc, aldst, N);
  k_ae<<<1, NHEAD, 0, stream>>>(We1, aed1, aeb);
  k_fill<<<(nh + 255) / 256, 256, 0, stream>>>(mbuf, -1e30f, nh);
  k_fill<<<(nh + 255) / 256, 256, 0, stream>>>(dbuf, 0.f, nh);
  k_fill<<<(nch + 255) / 256, 256, 0, stream>>>(agg, 0.f, nch);
  k_edge_max<<<(eh + 255) / 256, 256, 0, stream>>>(ei, alsrc, aldst, ew, aeb, alpha, mbuf, E);
  k_edge_agg<<<(E + EPB - 1) / EPB, 256, 0, stream>>>(ei, alpha, mbuf, h_buf, dbuf, agg, E);
  k_epilogue<<<N, 256, 0, stream>>>(agg, dbuf, b1, r_buf, g1, be1, r_buf, 1);

  // ---------------- Layer 2 ----------------
  k_gemm_f32wmma<<<gemmGrid, 256, 0, stream>>>(r_buf, W2, nullptr, h_buf, N, HID);
  k_node_alpha<<<(nh + 255) / 256, 256, 0, stream>>>(h_buf, asrc2, adst2, alsrc, aldst, N);
  k_ae<<<1, NHEAD, 0, stream>>>(We2, aed2, aeb);
  k_fill<<<(nh + 255) / 256, 256, 0, stream>>>(mbuf, -1e30f, nh);
  k_fill<<<(nh + 255) / 256, 256, 0, stream>>>(dbuf, 0.f, nh);
  k_fill<<<(nch + 255) / 256, 256, 0, stream>>>(agg, 0.f, nch);
  k_edge_max<<<(eh + 255) / 256, 256, 0, stream>>>(ei, alsrc, aldst, ew, aeb, alpha, mbuf, E);
  k_edge_agg<<<(E + EPB - 1) / EPB, 256, 0, stream>>>(ei, alpha, mbuf, h_buf, dbuf, agg, E);
  k_epilogue<<<N, 256, 0, stream>>>(agg, dbuf, b2, r_buf, g2, be2, (float*)d_out, 0);

  (void)n_in; (void)out_size; (void)ws_size;
}